// ULCDF_EXTRACTOR_6158983102642
// MI455X (gfx1250) — compile-verified
//
#include <hip/hip_runtime.h>
#include <hip/hip_bf16.h>
#include <cstdint>

typedef __attribute__((ext_vector_type(2))) float v2f;
typedef __attribute__((ext_vector_type(8))) float v8f;

#define SS   50000
#define EE   20000
#define KK   1000
#define DD   64
#define NNODE 71000
#define NNZC 2000000
#define BB   4096

// ---------------- utility kernels ----------------

__global__ void ulcdf_zero_i32(int* __restrict__ p, int n) {
  int i = blockIdx.x * blockDim.x + threadIdx.x;
  if (i < n) p[i] = 0;
}

__global__ void ulcdf_hist(const int* __restrict__ rows, int* __restrict__ cnt, int nnz) {
  int i = blockIdx.x * blockDim.x + threadIdx.x;
  if (i < nnz) atomicAdd(&cnt[rows[i]], 1);
}

// Single-block chunked exclusive scan over n entries; start[n] = total.
__global__ void ulcdf_scan(const int* __restrict__ cnt, int* __restrict__ start,
                           int* __restrict__ fill, int n) {
  __shared__ int sdata[1024];
  __shared__ int carry;
  if (threadIdx.x == 0) carry = 0;
  __syncthreads();
  for (int base = 0; base < n; base += 1024) {
    int i = base + (int)threadIdx.x;
    int v = (i < n) ? cnt[i] : 0;
    sdata[threadIdx.x] = v;
    __syncthreads();
    for (int off = 1; off < 1024; off <<= 1) {
      int t = (threadIdx.x >= (unsigned)off) ? sdata[threadIdx.x - off] : 0;
      __syncthreads();
      sdata[threadIdx.x] += t;
      __syncthreads();
    }
    int excl = sdata[threadIdx.x] - v + carry;
    if (i < n) { start[i] = excl; fill[i] = excl; }
    __syncthreads();
    if (threadIdx.x == 1023) carry += sdata[1023];
    __syncthreads();
  }
  if (threadIdx.x == 0) start[n] = carry;
}

__global__ void ulcdf_scatter(const int* __restrict__ rows, const int* __restrict__ cols,
                              const float* __restrict__ vals, int* __restrict__ fill,
                              int* __restrict__ colS, float* __restrict__ valS, int nnz) {
  int i = blockIdx.x * blockDim.x + threadIdx.x;
  if (i < nnz) {
    int r = rows[i];
    int pos = atomicAdd(&fill[r], 1);
    colS[pos] = cols[i];
    valS[pos] = vals[i];
  }
}

// x = acc = concat([stu, exer, know], axis=0)   (float4-vectorized; DD=64 -> 16 f4/row)
__global__ void ulcdf_init_emb(const float4* __restrict__ stu, const float4* __restrict__ exer,
                               const float4* __restrict__ know, float4* __restrict__ x,
                               float4* __restrict__ acc) {
  int i = blockIdx.x * blockDim.x + threadIdx.x;      // float4 index
  if (i < NNODE * (DD / 4)) {
    int row = i >> 4;
    float4 v;
    if (row < SS)           v = stu[i];
    else if (row < SS + EE) v = exer[i - SS * (DD / 4)];
    else                    v = know[i - (SS + EE) * (DD / 4)];
    x[i] = v;
    acc[i] = v;
  }
}

// Atomic-free CSR SpMM: one wave per row, lane holds dims {2l, 2l+1}. 2x unrolled gathers.
__global__ void ulcdf_spmm(const int* __restrict__ start, const int* __restrict__ colS,
                           const float* __restrict__ valS, const float* __restrict__ x,
                           float* __restrict__ y) {
  int wave = (int)((blockIdx.x * blockDim.x + threadIdx.x) >> 5);
  int lane = threadIdx.x & 31;
  if (wave >= NNODE) return;
  int s = start[wave];
  int e = start[wave + 1];
  float a0 = 0.0f, a1 = 0.0f;
  float b0 = 0.0f, b1 = 0.0f;
  int i = s;
  for (; i + 2 <= e; i += 2) {
    int   c0 = colS[i];
    int   c1 = colS[i + 1];
    float v0 = valS[i];
    float v1 = valS[i + 1];
    const float2 xv0 = *(const float2*)(x + ((size_t)c0 << 6) + (lane << 1));
    const float2 xv1 = *(const float2*)(x + ((size_t)c1 << 6) + (lane << 1));
    a0 = fmaf(v0, xv0.x, a0);
    a1 = fmaf(v0, xv0.y, a1);
    b0 = fmaf(v1, xv1.x, b0);
    b1 = fmaf(v1, xv1.y, b1);
  }
  if (i < e) {
    int   c0 = colS[i];
    float v0 = valS[i];
    const float2 xv0 = *(const float2*)(x + ((size_t)c0 << 6) + (lane << 1));
    a0 = fmaf(v0, xv0.x, a0);
    a1 = fmaf(v0, xv0.y, a1);
  }
  *(float2*)(y + ((size_t)wave << 6) + (lane << 1)) = make_float2(a0 + b0, a1 + b1);
}

// acc = (acc + y) * scale   (scale=0.25 on last layer folds the /(LAYERS+1))
__global__ void ulcdf_acc_add(float4* __restrict__ acc, const float4* __restrict__ y, float scale) {
  int i = blockIdx.x * blockDim.x + threadIdx.x;
  if (i < NNODE * (DD / 4)) {
    float4 a = acc[i];
    float4 b = y[i];
    a.x = (a.x + b.x) * scale;
    a.y = (a.y + b.y) * scale;
    a.z = (a.z + b.z) * scale;
    a.w = (a.w + b.w) * scale;
    acc[i] = a;
  }
}

// ---------------- WMMA f32 GEMM ----------------
// Out[M x NOUT] = leaky?( [A1 | A2][rows] @ W + bias )
// A1/A2: row-major, 64 floats/row. Compile-time NOUT -> immediate-offset loads.
// One wave computes a 16x16 tile via V_WMMA_F32_16X16X4_F32.
template <int NOUT, bool HASA2, bool GATHER, bool LEAKY>
__global__ void ulcdf_wmma_gemm(const float* __restrict__ A1, const float* __restrict__ A2,
                                const int* __restrict__ gather, int gatherOff,
                                const float* __restrict__ W, const float* __restrict__ bias,
                                float* __restrict__ Out, int M) {
  const int lane = threadIdx.x & 31;
  const int wv   = threadIdx.x >> 5;                 // 0..3
  const int nBase = (blockIdx.x * 4 + wv) * 16;
  const int mBase = blockIdx.y * 16;
  if (nBase >= NOUT) return;                         // wave-uniform exit

  const int lo = lane & 15;
  const int hi = lane >> 4;

  int mrow   = mBase + lo;
  int mclamp = (mrow < M) ? mrow : (M - 1);
  int arow   = GATHER ? (gather[mclamp] + gatherOff) : mclamp;

  int ncol   = nBase + lo;
  int nclamp = (ncol < NOUT) ? ncol : (NOUT - 1);

  // Per-wave invariant base pointers; K loop uses immediate offsets only.
  const float* a1p = A1 + (size_t)arow * 64 + 2 * hi;
  const float* wp  = W + (size_t)(2 * hi) * NOUT + nclamp;

  v8f c;
  float binit = bias[nclamp];
#pragma unroll
  for (int r = 0; r < 8; ++r) c[r] = binit;

  // A frag (16x4 f32): lanes 0-15 -> K {k,k+1}; lanes 16-31 -> K {k+2,k+3}; M = lane&15
  // B frag (4x16 f32): lane n = col n; VGPR0 = row k(+2 for hi half), VGPR1 = row k+1(+2)
#pragma unroll
  for (int k = 0; k < 64; k += 4) {
    v2f a = *(const v2f*)(a1p + k);
    v2f b;
    b.x = wp[k * NOUT];
    b.y = wp[k * NOUT + NOUT];
    c = __builtin_amdgcn_wmma_f32_16x16x4_f32(false, a, false, b, (short)0, c, false, false);
  }
  if (HASA2) {
    const float* a2p = A2 + (size_t)arow * 64 + 2 * hi;
    const float* wp2 = wp + (size_t)64 * NOUT;
#pragma unroll
    for (int k = 0; k < 64; k += 4) {
      v2f a = *(const v2f*)(a2p + k);
      v2f b;
      b.x = wp2[k * NOUT];
      b.y = wp2[k * NOUT + NOUT];
      c = __builtin_amdgcn_wmma_f32_16x16x4_f32(false, a, false, b, (short)0, c, false, false);
    }
  }

  // C/D layout: VGPR r, lanes 0-15 -> (M=r, N=lane); lanes 16-31 -> (M=r+8, N=lane-16)
  if (ncol < NOUT) {
#pragma unroll
    for (int r = 0; r < 8; ++r) {
      int row = mBase + r + 8 * hi;
      if (row < M) {
        float vv = c[r];
        if (LEAKY) vv = (vv >= 0.0f) ? vv : 0.8f * vv;
        Out[(size_t)row * NOUT + ncol] = vv;
      }
    }
  }
}

__global__ void ulcdf_disc(const float* __restrict__ disc_emb, const int* __restrict__ exid,
                           float* __restrict__ out) {
  int i = blockIdx.x * blockDim.x + threadIdx.x;
  if (i < BB) out[i] = disc_emb[exid[i]];
}

// ---------------- host ----------------

extern "C" void kernel_launch(void* const* d_in, const int* in_sizes, int n_in,
                              void* d_out, int out_size, void* d_ws, size_t ws_size,
                              hipStream_t stream) {
  (void)in_sizes; (void)n_in; (void)out_size; (void)ws_size;

  const float* stu       = (const float*)d_in[0];
  const float* exer      = (const float*)d_in[1];
  const float* know      = (const float*)d_in[2];
  const float* disc_emb  = (const float*)d_in[3];
  const int*   rowsA[3]  = {(const int*)d_in[4], (const int*)d_in[7], (const int*)d_in[10]};
  const int*   colsA[3]  = {(const int*)d_in[5], (const int*)d_in[8], (const int*)d_in[11]};
  const float* valsA[3]  = {(const float*)d_in[6], (const float*)d_in[9], (const float*)d_in[12]};
  const float* concat_W  = (const float*)d_in[13];
  const float* concat_b  = (const float*)d_in[14];
  const float* concat1_W = (const float*)d_in[15];
  const float* concat1_b = (const float*)d_in[16];
  const float* tstu_W    = (const float*)d_in[17];
  const float* tstu_b    = (const float*)d_in[18];
  const float* texer_W   = (const float*)d_in[19];
  const float* texer_b   = (const float*)d_in[20];
  const float* tknow_W   = (const float*)d_in[21];
  const float* tknow_b   = (const float*)d_in[22];
  const int*   student_id  = (const int*)d_in[23];
  const int*   exercise_id = (const int*)d_in[24];

  // workspace carve-out (~108 MB)
  char* ws = (char*)d_ws;
  size_t off = 0;
  auto walloc = [&](size_t bytes) -> void* {
    void* p = ws + off;
    off += (bytes + 255) & ~(size_t)255;
    return p;
  };
  const size_t embBytes = (size_t)NNODE * DD * sizeof(float);
  float* accb[3];                                 // 0=hol, 1=right, 2=wrong
  for (int g = 0; g < 3; ++g) accb[g] = (float*)walloc(embBytes);
  float* xbuf = (float*)walloc(embBytes);
  float* ybuf = (float*)walloc(embBytes);
  int*   cnt    = (int*)walloc((NNODE + 1) * sizeof(int));
  int*   startp = (int*)walloc((NNODE + 1) * sizeof(int));
  int*   fillp  = (int*)walloc((NNODE + 1) * sizeof(int));
  int*   colS   = (int*)walloc((size_t)NNZC * sizeof(int));
  float* valS   = (float*)walloc((size_t)NNZC * sizeof(float));

  const int T = 256;
  const int nVec4 = NNODE * (DD / 4);

  for (int g = 0; g < 3; ++g) {
    // counting sort of graph g by row (once; reused across 3 layers)
    ulcdf_zero_i32<<<(NNODE + 1 + T - 1) / T, T, 0, stream>>>(cnt, NNODE + 1);
    ulcdf_hist<<<(NNZC + T - 1) / T, T, 0, stream>>>(rowsA[g], cnt, NNZC);
    ulcdf_scan<<<1, 1024, 0, stream>>>(cnt, startp, fillp, NNODE);
    ulcdf_scatter<<<(NNZC + T - 1) / T, T, 0, stream>>>(rowsA[g], colsA[g], valsA[g],
                                                        fillp, colS, valS, NNZC);
    ulcdf_init_emb<<<(nVec4 + T - 1) / T, T, 0, stream>>>((const float4*)stu, (const float4*)exer,
                                                          (const float4*)know, (float4*)xbuf,
                                                          (float4*)accb[g]);
    float* xp = xbuf;
    float* yp = ybuf;
    for (int layer = 0; layer < 3; ++layer) {
      long long threads = (long long)NNODE * 32;
      ulcdf_spmm<<<(unsigned)((threads + T - 1) / T), T, 0, stream>>>(startp, colS, valS, xp, yp);
      float sc = (layer == 2) ? 0.25f : 1.0f;   // fold /(LAYERS+1) into last add
      ulcdf_acc_add<<<(nVec4 + T - 1) / T, T, 0, stream>>>((float4*)accb[g], (const float4*)yp, sc);
      float* t = xp; xp = yp; yp = t;
    }
  }

  float* dis  = xbuf;   // conv ping-pong buffers are free now
  float* outb = ybuf;

  dim3 blk(128);
  dim3 g1(1, (NNODE + 15) / 16);   // NOUT=64 -> 4 n-tiles = 4 waves of one block
  // dis = leaky([right | wrong] @ concat_W + concat_b)
  ulcdf_wmma_gemm<64, true, false, true><<<g1, blk, 0, stream>>>(
      accb[1], accb[2], nullptr, 0, concat_W, concat_b, dis, NNODE);
  // out = leaky([dis | hol] @ concat1_W + concat1_b)
  ulcdf_wmma_gemm<64, true, false, true><<<g1, blk, 0, stream>>>(
      dis, accb[0], nullptr, 0, concat1_W, concat1_b, outb, NNODE);

  float* out_stus  = (float*)d_out;                  // 4096 x 1000
  float* out_exers = (float*)d_out + 4096000;        // 4096 x 1000
  float* out_disc  = (float*)d_out + 8192000;        // 4096
  float* out_knows = (float*)d_out + 8196096;        // 1000 x 1000

  dim3 g2((1000 + 63) / 64, (BB + 15) / 16);
  ulcdf_wmma_gemm<1000, false, true, false><<<g2, blk, 0, stream>>>(
      outb, nullptr, student_id, 0, tstu_W, tstu_b, out_stus, BB);
  ulcdf_wmma_gemm<1000, false, true, false><<<g2, blk, 0, stream>>>(
      outb, nullptr, exercise_id, SS, texer_W, texer_b, out_exers, BB);
  dim3 g3((1000 + 63) / 64, (KK + 15) / 16);
  ulcdf_wmma_gemm<1000, false, false, false><<<g3, blk, 0, stream>>>(
      outb + (size_t)(SS + EE) * DD, nullptr, nullptr, 0, tknow_W, tknow_b, out_knows, KK);

  ulcdf_disc<<<(BB + T - 1) / T, T, 0, stream>>>(disc_emb, exercise_id, out_disc);
}